// AltupRouter_57784490000784
// MI455X (gfx1250) — compile-verified
//
#include <hip/hip_runtime.h>
#include <cmath>

#define DIM   2048
#define NEXP  4
#define EPSV  1e-6f
#define SCL   3.0f

typedef __attribute__((ext_vector_type(2))) float v2f;
typedef __attribute__((ext_vector_type(8))) float v8f;

// One wave (32 lanes) handles 16 rows. WMMA f32 16x16x4:
//   A (16x4): lane L holds row L%16, K = 2*(L>>4) + {0,1}  -> per-lane b64 load
//   B (4x16): lane L holds col N=L%16, K = 2*(L>>4) + {0,1} -> b64 from LDS
//   D (16x16): VGPR i: lanes 0-15 -> (M=i, N=lane), lanes 16-31 -> (M=8+i, N=lane-16)
__launch_bounds__(256, 2)
__global__ void altup_router_kernel(const float* __restrict__ x,
                                    const float* __restrict__ nw,
                                    const float* __restrict__ rw,
                                    float* __restrict__ out,
                                    int rows) {
  // rows 0..3: norm_weight * router_weight[e]; row 4: zeros (for unused B columns)
  __shared__ float wlds[5 * DIM];

  const int tid = threadIdx.x;
  for (int t = tid; t < 5 * DIM; t += 256) {
    const int e = t >> 11;                       // t / DIM
    wlds[t] = (e < NEXP) ? nw[t & (DIM - 1)] * rw[t] : 0.0f;
  }
  __syncthreads();

  const int lane    = tid & 31;
  const int wave    = tid >> 5;
  const int e       = lane & 15;                 // column (expert id for e<4)
  const int half    = lane >> 4;                 // lane half selects K sub-slot
  const int rowTile = blockIdx.x * 128 + wave * 16;
  if (rowTile >= rows) return;
  const int row = rowTile + e;                   // row this lane loads (A matrix)

  const float* px = x + (size_t)row * DIM + (half << 1);
  const int    wb = ((e < NEXP) ? e : NEXP) * DIM + (half << 1);

  v8f c   = {};
  v2f ss0 = {0.0f, 0.0f};                        // packed sumsq accumulators
  v2f ss1 = {0.0f, 0.0f};                        // (two chains -> v_pk_fma_f32)

#pragma unroll 4
  for (int k = 0; k < DIM; k += 16) {
    v2f a0 = *(const v2f*)(px + k);
    v2f a1 = *(const v2f*)(px + k + 4);
    v2f a2 = *(const v2f*)(px + k + 8);
    v2f a3 = *(const v2f*)(px + k + 12);
    v2f b0 = *(const v2f*)&wlds[wb + k];
    v2f b1 = *(const v2f*)&wlds[wb + k + 4];
    v2f b2 = *(const v2f*)&wlds[wb + k + 8];
    v2f b3 = *(const v2f*)&wlds[wb + k + 12];

    c = __builtin_amdgcn_wmma_f32_16x16x4_f32(false, a0, false, b0, (short)0, c, false, false);
    c = __builtin_amdgcn_wmma_f32_16x16x4_f32(false, a1, false, b1, (short)0, c, false, false);
    c = __builtin_amdgcn_wmma_f32_16x16x4_f32(false, a2, false, b2, (short)0, c, false, false);
    c = __builtin_amdgcn_wmma_f32_16x16x4_f32(false, a3, false, b3, (short)0, c, false, false);

    ss0 += a0 * a0;
    ss1 += a1 * a1;
    ss0 += a2 * a2;
    ss1 += a3 * a3;
  }

  v2f ssv = ss0 + ss1;
  float sumsq = ssv.x + ssv.y;

  // Row sumsq lives split across lane L and lane L+16 -> combine halves.
  sumsq += __shfl_xor(sumsq, 16);
  const float inv = rsqrtf(sumsq * (1.0f / DIM) + EPSV);   // 1/rms for row (lane&15)

  // Lane's c[i] belongs to row (half*8 + i); fetch that row's inv-rms from lane half*8+i.
  const int srcbase = half << 3;
#pragma unroll
  for (int i = 0; i < 8; ++i) {
    const float inv_i = __shfl(inv, srcbase + i);
    const float val   = tanhf(SCL * c[i] * inv_i);
    if (e < NEXP)
      out[(size_t)(rowTile + srcbase + i) * NEXP + e] = val;
  }
}

extern "C" void kernel_launch(void* const* d_in, const int* in_sizes, int n_in,
                              void* d_out, int out_size, void* d_ws, size_t ws_size,
                              hipStream_t stream) {
  const float* x  = (const float*)d_in[0];   // (4, 8192, 2048) f32
  const float* nw = (const float*)d_in[1];   // (2048,) f32
  const float* rw = (const float*)d_in[2];   // (4, 2048) f32
  float* out = (float*)d_out;                // (4, 8192, 4) f32

  const int rows   = in_sizes[0] / DIM;      // 32768
  const int blocks = (rows + 127) / 128;     // 128 rows per 256-thread block
  altup_router_kernel<<<blocks, 256, 0, stream>>>(x, nw, rw, out, rows);
}